// SSIMLoss_80839874445893
// MI455X (gfx1250) — compile-verified
//
#include <hip/hip_runtime.h>
#include <hip/hip_bf16.h>

typedef __attribute__((ext_vector_type(2))) float v2f;
typedef __attribute__((ext_vector_type(8))) float v8f;

#define IMGW    512
#define TILE    16
#define OTROWS  64          // output rows per block (x 16 cols)
#define RR      80          // region rows, 74 valid, padded to 5 M-tiles
#define RVALID  74          // OTROWS + 10 halo rows
#define RCOLS   28          // region cols, 26 valid, K padded to 28
#define TSTRIDE 17          // tmp row stride (bank-conflict-free)
#define KSTEPS  7           // 28 / 4

__device__ __forceinline__ v8f wmma4(v2f a, v2f b, v8f c) {
    // D = A(16x4,f32) * B(4x16,f32) + C(16x16,f32)
    return __builtin_amdgcn_wmma_f32_16x16x4_f32(false, a, false, b, (short)0, c,
                                                 false, false);
}

__global__ __launch_bounds__(128) void ssim_tile_kernel(
        const float* __restrict__ img1,
        const float* __restrict__ img2,
        double* __restrict__ accum)
{
    __shared__ float w_s[12];                       // [11] = 0 pad for band OOB
    __shared__ __align__(16) float reg1[RR * RCOLS];
    __shared__ __align__(16) float reg2[RR * RCOLS];
    __shared__ float tmp0[RR * TSTRIDE];
    __shared__ float tmp1[RR * TSTRIDE];
    __shared__ float tmp2[RR * TSTRIDE];
    __shared__ float tmp3[RR * TSTRIDE];
    __shared__ float tmp4[RR * TSTRIDE];
    __shared__ float partial[4];

    const int tid = threadIdx.x;

    // Normalized 11-tap Gaussian window (sigma = 1.5), fp32 like the reference.
    float s = 0.f;
#pragma unroll
    for (int i = 0; i < 11; ++i) {
        float k = (float)(i - 5);
        s += expf(-0.5f / (1.5f * 1.5f) * k * k);
    }
    if (tid < 11) {
        float k = (float)(tid - 5);
        w_s[tid] = expf(-0.5f / (1.5f * 1.5f) * k * k) / s;
    }
    if (tid == 11) w_s[11] = 0.f;

    // Tile coordinates: 8 x 32 tiles of 64x16 per 512x512 plane, 96 planes.
    const int plane = blockIdx.x >> 8;
    const int t     = blockIdx.x & 255;
    const int y0    = (t >> 5) << 6;                // 64-row strips
    const int x0    = (t & 31) << 4;                // 16-col strips
    const size_t base = (size_t)plane * (IMGW * IMGW);

    // Load 74x26 halo region, zero-padded to 80x28 (image borders + K/M pad;
    // padded rows/cols propagate zeros, never NaN).
    for (int idx = tid; idx < RR * RCOLS; idx += 128) {
        int r = idx / RCOLS;
        int c = idx - r * RCOLS;
        int gy = y0 + r - 5;
        int gx = x0 + c - 5;
        float v1 = 0.f, v2 = 0.f;
        if (r < RVALID && c < 26 && (unsigned)gy < (unsigned)IMGW &&
            (unsigned)gx < (unsigned)IMGW) {
            size_t o = base + (size_t)gy * IMGW + gx;
            v1 = img1[o];
            v2 = img2[o];
        }
        reg1[idx] = v1;
        reg2[idx] = v2;
    }
    __syncthreads();

    const int wave    = tid >> 5;
    const int lane    = tid & 31;
    const int halfsel = lane >> 4;   // selects K pair {0,1} vs {2,3}
    const int mn      = lane & 15;   // M (A rows) or N (B cols)

    // Hoisted band fragments: identical for horizontal-B and vertical-A.
    // Band[j][n] = w[j-n] (resp. A[m][j] = w[j-m]); slot 11 of w_s is 0.
#define BANDW(d) w_s[((unsigned)(d) <= 10u) ? (d) : 11]
    v2f bw[KSTEPS];
#pragma unroll
    for (int k = 0; k < KSTEPS; ++k) {
        const int j0 = 4 * k + 2 * halfsel;
        bw[k].x = BANDW(j0 - mn);
        bw[k].y = BANDW(j0 + 1 - mn);
    }
#undef BANDW

    // ---------------- horizontal pass ----------------
    // tmp_i(80x16) = region_i(80x28) x Band(28x16); M-tiles striped on waves.
    for (int mt = wave; mt < RR / 16; mt += 4) {
        const int row = mt * 16 + mn;       // A-fragment row for this lane
        v8f a0 = {}, a1 = {}, a2 = {}, a3 = {}, a4 = {};
#pragma unroll
        for (int k = 0; k < KSTEPS; ++k) {
            const int j0 = 4 * k + 2 * halfsel;
            const v2f r1 = *(const v2f*)&reg1[row * RCOLS + j0];
            const v2f r2 = *(const v2f*)&reg2[row * RCOLS + j0];
            v2f p11 = { r1.x * r1.x, r1.y * r1.y };
            v2f p22 = { r2.x * r2.x, r2.y * r2.y };
            v2f p12 = { r1.x * r2.x, r1.y * r2.y };
            a0 = wmma4(r1,  bw[k], a0);
            a1 = wmma4(r2,  bw[k], a1);
            a2 = wmma4(p11, bw[k], a2);
            a3 = wmma4(p22, bw[k], a3);
            a4 = wmma4(p12, bw[k], a4);
        }
        // D layout: VGPR v holds D[v + 8*halfsel][mn]
        const int orow = mt * 16 + 8 * halfsel;
#pragma unroll
        for (int v = 0; v < 8; ++v) {
            tmp0[(orow + v) * TSTRIDE + mn] = a0[v];
            tmp1[(orow + v) * TSTRIDE + mn] = a1[v];
            tmp2[(orow + v) * TSTRIDE + mn] = a2[v];
            tmp3[(orow + v) * TSTRIDE + mn] = a3[v];
            tmp4[(orow + v) * TSTRIDE + mn] = a4[v];
        }
    }
    __syncthreads();

    // ---------------- vertical pass ----------------
    // One 16x16 output tile per wave: Out = Band(16x28) x tmp[rb..rb+27](28x16)
    const int rb = wave * 16;               // tmp row base for this out-tile
    v8f m1 = {}, m2 = {}, q11 = {}, q22 = {}, q12 = {};
#pragma unroll
    for (int k = 0; k < KSTEPS; ++k) {
        const int j0 = rb + 4 * k + 2 * halfsel;
        v2f b0 = { tmp0[j0 * TSTRIDE + mn], tmp0[(j0 + 1) * TSTRIDE + mn] };
        v2f b1 = { tmp1[j0 * TSTRIDE + mn], tmp1[(j0 + 1) * TSTRIDE + mn] };
        v2f b2 = { tmp2[j0 * TSTRIDE + mn], tmp2[(j0 + 1) * TSTRIDE + mn] };
        v2f b3 = { tmp3[j0 * TSTRIDE + mn], tmp3[(j0 + 1) * TSTRIDE + mn] };
        v2f b4 = { tmp4[j0 * TSTRIDE + mn], tmp4[(j0 + 1) * TSTRIDE + mn] };
        m1  = wmma4(bw[k], b0, m1);
        m2  = wmma4(bw[k], b1, m2);
        q11 = wmma4(bw[k], b2, q11);
        q22 = wmma4(bw[k], b3, q22);
        q12 = wmma4(bw[k], b4, q12);
    }

    // ---------------- SSIM map + reduction ----------------
    const float C1 = 0.0001f;   // (0.01)^2
    const float C2 = 0.0009f;   // (0.03)^2
    float lsum = 0.f;
#pragma unroll
    for (int v = 0; v < 8; ++v) {
        float mu1 = m1[v], mu2 = m2[v];
        float mu1s = mu1 * mu1, mu2s = mu2 * mu2, mu12 = mu1 * mu2;
        float sg1  = q11[v] - mu1s;
        float sg2  = q22[v] - mu2s;
        float sg12 = q12[v] - mu12;
        float num = (2.f * mu12 + C1) * (2.f * sg12 + C2);
        float den = (mu1s + mu2s + C1) * (sg1 + sg2 + C2);
        lsum += num / den;
    }
#pragma unroll
    for (int off = 16; off > 0; off >>= 1)
        lsum += __shfl_xor(lsum, off, 32);
    if (lane == 0) partial[wave] = lsum;
    __syncthreads();
    if (tid == 0) {
        float bsum = partial[0] + partial[1] + partial[2] + partial[3];
        atomicAdd(accum, (double)bsum);
    }
}

__global__ void ssim_init_kernel(double* a) { *a = 0.0; }

__global__ void ssim_final_kernel(const double* __restrict__ a,
                                  float* __restrict__ out) {
    const double N = 32.0 * 3.0 * 512.0 * 512.0;
    *out = 1.0f - (float)(*a / N);
}

extern "C" void kernel_launch(void* const* d_in, const int* in_sizes, int n_in,
                              void* d_out, int out_size, void* d_ws, size_t ws_size,
                              hipStream_t stream) {
    (void)in_sizes; (void)n_in; (void)out_size; (void)ws_size;
    const float* img1 = (const float*)d_in[0];
    const float* img2 = (const float*)d_in[1];
    double* acc = (double*)d_ws;   // 8 bytes of scratch for the global sum

    ssim_init_kernel<<<1, 1, 0, stream>>>(acc);
    // 96 planes * (8x32) tiles of 64x16 outputs, 4 waves (128 threads) / block
    ssim_tile_kernel<<<96 * 256, 128, 0, stream>>>(img1, img2, acc);
    ssim_final_kernel<<<1, 1, 0, stream>>>(acc, (float*)d_out);
}